// DiseaseLevelShapleyGame_34445637714521
// MI455X (gfx1250) — compile-verified
//
#include <hip/hip_runtime.h>

#define B_SZ 16384
#define D_SZ 4096
#define K_SZ 14
#define EPSN 1e-12f
#define TSTR 18  // padded LDS row stride (floats) -> avoids bank conflicts, keeps 8B align

typedef __attribute__((ext_vector_type(2))) float v2f;
typedef __attribute__((ext_vector_type(8))) float v8f;

// workspace layout (floats)
#define WS_PPAD 0                           // 16 x 4096 normalized, zero-padded prototypes
#define WS_VSIM (16 * D_SZ)                 // 2 x B x 14 branch similarities
#define WS_PAIR (WS_VSIM + 2 * B_SZ * K_SZ) // 196 per-pair loss terms

// ---------------- kernel 1: normalize prototypes (padded to 16 rows) + copy raw protos to out
__global__ void proto_norm_kernel(const float* __restrict__ p,
                                  float* __restrict__ ppad,
                                  float* __restrict__ out_proto) {
  __shared__ float red[256];
  const int row = blockIdx.x;     // 0..15
  const int tid = threadIdx.x;
  if (row >= K_SZ) {              // pad rows 14,15 with zeros
    for (int j = tid; j < D_SZ; j += 256) ppad[row * D_SZ + j] = 0.0f;
    return;
  }
  float s = 0.0f;
  for (int j = tid; j < D_SZ; j += 256) {
    float x = p[row * D_SZ + j];
    s = fmaf(x, x, s);
  }
  red[tid] = s;
  __syncthreads();
  for (int st = 128; st > 0; st >>= 1) {
    if (tid < st) red[tid] += red[tid + st];
    __syncthreads();
  }
  const float inv = 1.0f / fmaxf(sqrtf(red[0]), EPSN);
  for (int j = tid; j < D_SZ; j += 256) {
    float x = p[row * D_SZ + j];
    ppad[row * D_SZ + j] = x * inv;
    out_proto[row * D_SZ + j] = x;   // reference returns raw prototypes
  }
}

// ---------------- kernel 2: per-(i,j) co-occurrence loss terms
__global__ void cooccur_pair_kernel(const int* __restrict__ labels,
                                    const float* __restrict__ co,
                                    float* __restrict__ pair_out) {
  __shared__ float red[256];
  const int t = blockIdx.x;            // 0..195
  const int i = t / K_SZ, j = t % K_SZ;
  const int tid = threadIdx.x;
  int acc = 0;
  for (int b = tid; b < B_SZ; b += 256)
    acc += labels[b * K_SZ + i] * labels[b * K_SZ + j];
  red[tid] = (float)acc;
  __syncthreads();
  for (int st = 128; st > 0; st >>= 1) {
    if (tid < st) red[tid] += red[tid + st];
    __syncthreads();
  }
  if (tid == 0) {
    float bc = red[0] * (1.0f / (float)B_SZ);
    float sg = 1.0f / (1.0f + __expf(-co[t]));
    float d = sg - bc;
    pair_out[t] = d * d;
  }
}

// ---------------- kernel 3: mean of 196 terms -> scalar loss
__global__ void loss_final_kernel(const float* __restrict__ pair_in,
                                  float* __restrict__ out_scalar) {
  __shared__ float red[256];
  const int tid = threadIdx.x;
  red[tid] = (tid < K_SZ * K_SZ) ? pair_in[tid] : 0.0f;
  __syncthreads();
  for (int st = 128; st > 0; st >>= 1) {
    if (tid < st) red[tid] += red[tid + st];
    __syncthreads();
  }
  if (tid == 0) out_scalar[0] = red[0] * (1.0f / (float)(K_SZ * K_SZ));
}

// ---------------- kernel 4: fused  X@W.T + b  ->  row norm + prototype similarities
// grid (B/32, 2 branches), 128 threads = 4 waves.
// Each wave: 32 rows (2 row-groups) x 1024 cols; B fragments reused across both row-groups.
__global__ __launch_bounds__(128) void fused_branch_kernel(
    const float* __restrict__ Ximg, const float* __restrict__ Xtxt,
    const float* __restrict__ Wv,   const float* __restrict__ Wt,
    const float* __restrict__ bv,   const float* __restrict__ bt,
    const float* __restrict__ ppad, float* __restrict__ vsim) {
  __shared__ float ytile[4 * 16 * TSTR];   // per-wave 16x16 Y tile (padded)
  __shared__ float red_s[4][32][16];       // per-wave partial proto sims
  __shared__ float red_n[4][32];           // per-wave partial row norms

  const int branch = blockIdx.y;
  const float* __restrict__ X    = branch ? Xtxt : Ximg;
  const float* __restrict__ W    = branch ? Wt   : Wv;
  const float* __restrict__ bias = branch ? bt   : bv;

  const int tid  = threadIdx.x;
  const int wave = tid >> 5;
  const int lane = tid & 31;
  const int half = lane >> 4;    // 0: K=0..1  1: K=2..3 of the 16x16x4 fragment
  const int lid  = lane & 15;

  const int row0 = blockIdx.x * 32;

  v8f s_acc[2];
  s_acc[0] = (v8f){};
  s_acc[1] = (v8f){};
  float nrm[2][8];
#pragma unroll
  for (int g = 0; g < 2; ++g)
#pragma unroll
    for (int r = 0; r < 8; ++r) nrm[g][r] = 0.0f;

  const float* Arow0 = X + (size_t)(row0 + lid) * D_SZ + 2 * half;
  const float* Arow1 = Arow0 + (size_t)16 * D_SZ;
  float* myY = &ytile[wave * 16 * TSTR];

  for (int c = 0; c < 16; ++c) {           // 16 chunks of 64 columns per wave
    const int jbase = wave * 1024 + c * 64;
    v8f acc[2][4];
#pragma unroll
    for (int g = 0; g < 2; ++g)
#pragma unroll
      for (int t = 0; t < 4; ++t) acc[g][t] = (v8f){};
    const float* Bp[4];
#pragma unroll
    for (int t = 0; t < 4; ++t)
      Bp[t] = W + (size_t)(jbase + t * 16 + lid) * D_SZ + 2 * half;

#pragma unroll 2
    for (int k = 0; k < D_SZ; k += 4) {    // the GEMM K loop: 6 loads -> 8 WMMAs
      v2f a0 = *(const v2f*)(Arow0 + k);
      v2f a1 = *(const v2f*)(Arow1 + k);
#pragma unroll
      for (int t = 0; t < 4; ++t) {
        v2f b = *(const v2f*)(Bp[t] + k);
        acc[0][t] = __builtin_amdgcn_wmma_f32_16x16x4_f32(
            false, a0, false, b, (short)0, acc[0][t], false, false);
        acc[1][t] = __builtin_amdgcn_wmma_f32_16x16x4_f32(
            false, a1, false, b, (short)0, acc[1][t], false, false);
      }
    }

    // epilogue per 16x16 tile: bias, norm accumulation, proto-sim WMMA via LDS
#pragma unroll
    for (int t = 0; t < 4; ++t) {
      const int jt = jbase + t * 16;
      const float bb = bias[jt + lid];     // column bias (N = lid for all lanes)
#pragma unroll
      for (int g = 0; g < 2; ++g) {
#pragma unroll
        for (int r = 0; r < 8; ++r) {
          float y = acc[g][t][r] + bb;     // C layout: row = r + 8*half, col = lid
          nrm[g][r] = fmaf(y, y, nrm[g][r]);
          myY[(r + 8 * half) * TSTR + lid] = y;
        }
        asm volatile("s_wait_dscnt 0" ::: "memory");  // wave-synchronous LDS handoff
#pragma unroll
        for (int ks = 0; ks < 4; ++ks) {
          // A frag: Y[m=lid, kk=ks*4+2*half+{0,1}]
          v2f ya = *(const v2f*)(&myY[lid * TSTR + ks * 4 + 2 * half]);
          // B frag: ppad[proto=lid][jt + ks*4 + 2*half + {0,1}] (rows 14,15 are zero)
          v2f pb = *(const v2f*)(&ppad[(size_t)lid * D_SZ + jt + ks * 4 + 2 * half]);
          s_acc[g] = __builtin_amdgcn_wmma_f32_16x16x4_f32(
              false, ya, false, pb, (short)0, s_acc[g], false, false);
        }
      }
    }
  }

  // deterministic cross-wave combine through LDS
#pragma unroll
  for (int g = 0; g < 2; ++g) {
#pragma unroll
    for (int r = 0; r < 8; ++r) {
      const int m = g * 16 + r + 8 * half;
      red_s[wave][m][lid] = s_acc[g][r];
      float v = nrm[g][r];
      v += __shfl_xor(v, 1);
      v += __shfl_xor(v, 2);
      v += __shfl_xor(v, 4);
      v += __shfl_xor(v, 8);               // sum over the 16 lanes of this half
      if (lid == 0) red_n[wave][m] = v;
    }
  }
  __syncthreads();

  for (int idx = tid; idx < 32 * K_SZ; idx += 128) {
    const int m = idx / K_SZ, kp = idx % K_SZ;
    float s = red_s[0][m][kp] + red_s[1][m][kp] + red_s[2][m][kp] + red_s[3][m][kp];
    float nn = red_n[0][m] + red_n[1][m] + red_n[2][m] + red_n[3][m];
    float sim = s / fmaxf(sqrtf(nn), EPSN);
    vsim[((size_t)branch * B_SZ + row0 + m) * K_SZ + kp] = sim;
  }
}

// ---------------- kernel 5: shapley = 0.5*(v_sim + t_sim) * labels
__global__ void shapley_kernel(const float* __restrict__ vsim,
                               const int* __restrict__ labels,
                               float* __restrict__ out) {
  const int i = blockIdx.x * 256 + threadIdx.x;
  if (i < B_SZ * K_SZ) {
    float v = vsim[i];
    float t = vsim[B_SZ * K_SZ + i];
    out[i] = 0.5f * (v + t) * (float)labels[i];
  }
}

extern "C" void kernel_launch(void* const* d_in, const int* in_sizes, int n_in,
                              void* d_out, int out_size, void* d_ws, size_t ws_size,
                              hipStream_t stream) {
  const float* image  = (const float*)d_in[0];
  const float* text   = (const float*)d_in[1];
  const int*   labels = (const int*)  d_in[2];
  const float* Wv     = (const float*)d_in[3];
  const float* bv     = (const float*)d_in[4];
  const float* Wt     = (const float*)d_in[5];
  const float* bt     = (const float*)d_in[6];
  const float* protos = (const float*)d_in[7];
  const float* co     = (const float*)d_in[8];

  float* out = (float*)d_out;
  float* out_shap  = out;                              // B*K
  float* out_proto = out + B_SZ * K_SZ;                // K*D
  float* out_loss  = out + B_SZ * K_SZ + K_SZ * D_SZ;  // scalar

  float* ws   = (float*)d_ws;
  float* ppad = ws + WS_PPAD;
  float* vsim = ws + WS_VSIM;
  float* pair = ws + WS_PAIR;

  proto_norm_kernel<<<16, 256, 0, stream>>>(protos, ppad, out_proto);
  cooccur_pair_kernel<<<K_SZ * K_SZ, 256, 0, stream>>>(labels, co, pair);
  loss_final_kernel<<<1, 256, 0, stream>>>(pair, out_loss);

  dim3 grid(B_SZ / 32, 2);
  fused_branch_kernel<<<grid, 128, 0, stream>>>(image, text, Wv, Wt, bv, bt, ppad, vsim);

  shapley_kernel<<<(B_SZ * K_SZ + 255) / 256, 256, 0, stream>>>(vsim, labels, out_shap);
}